// GCN_7241314861485
// MI455X (gfx1250) — compile-verified
//
#include <hip/hip_runtime.h>

typedef float v2f __attribute__((ext_vector_type(2)));
typedef float v8f __attribute__((ext_vector_type(8)));

// ---------------- degree / dinv ----------------
__global__ void k_deg_init(float* __restrict__ deg, int n) {
    int i = blockIdx.x * blockDim.x + threadIdx.x;
    if (i < n) deg[i] = 1.0f;  // self-loop contributes 1
}

__global__ void k_deg_acc(const int* __restrict__ dst, float* __restrict__ deg, int E) {
    int e = blockIdx.x * blockDim.x + threadIdx.x;
    if (e < E) atomicAdd(&deg[dst[e]], 1.0f);
}

__global__ void k_rsqrt_inplace(float* __restrict__ deg, int n) {
    int i = blockIdx.x * blockDim.x + threadIdx.x;
    if (i < n) deg[i] = rsqrtf(deg[i]);
}

// ---------------- WMMA f32 GEMM: C[M,N] = A[M,K] * B[K,N] ----------------
// One wave per 16x16 C tile. M multiple of 16, N,K compile-time (mult of 16 / 4).
// A 16x4 layout: lanes 0-15 -> rows, VGPR0/1 = K={0,1} (half 0) or K={2,3} (half 1).
// B 4x16 layout: lanes 0-15 -> cols, VGPR0/1 = K rows analogously.
// C 16x16 layout: lane&15 -> col, VGPR v -> row v (lanes 0-15) / v+8 (lanes 16-31).
template <int N, int K>
__global__ void k_gemm_wmma(const float* __restrict__ A, const float* __restrict__ B,
                            float* __restrict__ C) {
    const int wave  = (blockIdx.x * blockDim.x + threadIdx.x) >> 5;
    const int lane  = threadIdx.x & 31;
    const int tilesN = N / 16;
    const int tM   = wave / tilesN;
    const int tN   = wave % tilesN;
    const int half = lane >> 4;   // 0 or 1
    const int l    = lane & 15;

    const size_t rowA = (size_t)tM * 16 + l;
    const int    colB = tN * 16 + l;

    const float* ap = A + rowA * K + half * 2;

    v8f acc = {};
#pragma unroll
    for (int k = 0; k < K; k += 4) {
        v2f a = *(const v2f*)(ap + k);          // K = k+2h, k+2h+1 for this lane's row
        const int kb = k + half * 2;
        v2f b;
        b.x = B[(size_t)kb * N + colB];
        b.y = B[(size_t)(kb + 1) * N + colB];
        acc = __builtin_amdgcn_wmma_f32_16x16x4_f32(
            /*neg_a=*/false, a, /*neg_b=*/false, b,
            /*c_mod=*/(short)0, acc, /*reuse_a=*/false, /*reuse_b=*/false);
    }

    float* cp = C + ((size_t)tM * 16 + half * 8) * N + (size_t)tN * 16 + l;
#pragma unroll
    for (int v = 0; v < 8; ++v) cp[(size_t)v * N] = acc[v];
}

// ---------------- self-loop init: out[i,f] = h[i,f]*dinv[i]^2 (+ bias[f]) ----------------
template <int F, bool BIAS>
__global__ void k_self_init(const float* __restrict__ h, const float* __restrict__ dinv,
                            const float* __restrict__ bias, float* __restrict__ out, int n) {
    int t = blockIdx.x * blockDim.x + threadIdx.x;
    if (t >= n * F) return;
    int node = t / F;
    int f    = t % F;
    float di = dinv[node];
    float v  = h[t] * di * di;
    if (BIAS) v += bias[f];
    out[t] = v;
}

// ---------------- bias + relu: out[i,f] = max(agg[i,f]+bias[f], 0) ----------------
template <int F>
__global__ void k_bias_relu(const float* __restrict__ agg, const float* __restrict__ bias,
                            float* __restrict__ out, int n) {
    int t = blockIdx.x * blockDim.x + threadIdx.x;
    if (t >= n * F) return;
    int f = t % F;
    float v = agg[t] + bias[f];
    out[t] = v > 0.0f ? v : 0.0f;
}

// ---------------- edge scatter: agg[dst] += h[src] * dinv[src]*dinv[dst] ----------------
// One wave per edge; each lane handles F/32 contiguous floats (coalesced row).
template <int F>
__global__ void k_edge_scatter(const int* __restrict__ src, const int* __restrict__ dst,
                               const float* __restrict__ dinv, const float* __restrict__ h,
                               float* __restrict__ agg, int E) {
    int t = blockIdx.x * blockDim.x + threadIdx.x;
    int e = t >> 5;
    if (e >= E) return;
    int lane = t & 31;
    int s = src[e], d = dst[e];
    float nrm = dinv[s] * dinv[d];
    constexpr int PER = F / 32;
    const float* hp = h   + (size_t)s * F + lane * PER;
    float*       op = agg + (size_t)d * F + lane * PER;
#pragma unroll
    for (int j = 0; j < PER; ++j) atomicAdd(op + j, hp[j] * nrm);
}

extern "C" void kernel_launch(void* const* d_in, const int* in_sizes, int n_in,
                              void* d_out, int out_size, void* d_ws, size_t ws_size,
                              hipStream_t stream) {
    const float* x  = (const float*)d_in[0];
    const int*   ei = (const int*)d_in[1];
    const float* W1 = (const float*)d_in[2];
    const float* b1 = (const float*)d_in[3];
    const float* W2 = (const float*)d_in[4];
    const float* b2 = (const float*)d_in[5];
    float* out = (float*)d_out;

    const int n = in_sizes[0] / 128;   // 100000
    const int E = in_sizes[1] / 2;     // 1600000
    const int* src = ei;
    const int* dst = ei + E;

    // workspace layout
    char* ws = (char*)d_ws;
    size_t hOff   = ((size_t)n * 4 + 511) & ~(size_t)511;
    size_t aggOff = hOff + (size_t)n * 128 * 4;
    size_t gOff   = aggOff + (size_t)n * 128 * 4;
    float* dinv = (float*)(ws);          // n floats (deg -> dinv in place)
    float* h    = (float*)(ws + hOff);   // n*128
    float* agg  = (float*)(ws + aggOff); // n*128
    float* g    = (float*)(ws + gOff);   // n*64

    const int B = 256;

    // degree -> dinv
    k_deg_init<<<(n + B - 1) / B, B, 0, stream>>>(dinv, n);
    k_deg_acc<<<(E + B - 1) / B, B, 0, stream>>>(dst, dinv, E);
    k_rsqrt_inplace<<<(n + B - 1) / B, B, 0, stream>>>(dinv, n);

    // ----- conv1 -----
    {
        int waves = (n / 16) * (128 / 16);              // 50000 full waves
        k_gemm_wmma<128, 128><<<waves / 8, 256, 0, stream>>>(x, W1, h);
    }
    k_self_init<128, false><<<((size_t)n * 128 + B - 1) / B, B, 0, stream>>>(h, dinv, nullptr, agg, n);
    k_edge_scatter<128><<<((size_t)E * 32 + B - 1) / B, B, 0, stream>>>(src, dst, dinv, h, agg, E);
    k_bias_relu<128><<<((size_t)n * 128 + B - 1) / B, B, 0, stream>>>(agg, b1, h, n);  // h := relu(conv1)

    // ----- conv2 -----
    {
        int waves = (n / 16) * (64 / 16);               // 25000 full waves
        k_gemm_wmma<64, 128><<<waves / 8, 256, 0, stream>>>(h, W2, g);
    }
    k_self_init<64, true><<<((size_t)n * 64 + B - 1) / B, B, 0, stream>>>(g, dinv, b2, out, n);
    k_edge_scatter<64><<<((size_t)E * 32 + B - 1) / B, B, 0, stream>>>(src, dst, dinv, g, out, E);
}